// TemporalAttention_77790447665640
// MI455X (gfx1250) — compile-verified
//
#include <hip/hip_runtime.h>

typedef float v2f __attribute__((ext_vector_type(2)));
typedef float v4f __attribute__((ext_vector_type(4)));
typedef float v8f __attribute__((ext_vector_type(8)));

#define NB 8
#define NC 64
#define NSP 131072            // 32*64*64 spatial elements per (b,c)
#define NSP4 (NSP / 4)
#define K1_BLOCKS_PER_BATCH 128
#define K1_WAVES_PER_BATCH (K1_BLOCKS_PER_BATCH * 8)
#define TILES_PER_BATCH (NSP / 16)   // 8192, divisible by K1_WAVES_PER_BATCH

// ---------------------------------------------------------------------------
// Pass 1: per 16-spatial tile, compute [t;g] = W(2x64) * in(64x16) with
// V_WMMA_F32_16X16X4_F32 accumulating K=64 in 16 steps of K=4.
//   A (16x4): row0 = theta_w slice, row1 = g_w slice, rows 2..15 = 0.
//   B (4x16): 4 channels x 16 consecutive spatial positions.
//   D: vgpr0 lanes0-15 = t[n0..n0+15], vgpr1 lanes0-15 = g[n0..n0+15].
// g goes to the workspace; t is wave-reduced and atomically added per batch.
// ---------------------------------------------------------------------------
__global__ __launch_bounds__(256)
void k1_contract_wmma(const float* __restrict__ in,
                      const float* __restrict__ theta_w,
                      const float* __restrict__ g_w,
                      float* __restrict__ tsum,
                      float* __restrict__ gbuf)
{
    const int lane   = threadIdx.x & 31;
    const int wave   = threadIdx.x >> 5;
    const int b      = blockIdx.x / K1_BLOCKS_PER_BATCH;
    const int blk    = blockIdx.x % K1_BLOCKS_PER_BATCH;
    const int waveId = blk * 8 + wave;

    const int m     = lane & 15;   // A: row M; B/D: column N (spatial)
    const int khalf = lane >> 4;   // which K pair this half-wave carries

    // Precompute the 16 A operands (held in registers via full unroll).
    v2f A[16];
#pragma unroll
    for (int k = 0; k < 16; ++k) {
        const int c = 4 * k + 2 * khalf;
        float wx = 0.0f, wy = 0.0f;
        if (m == 0)      { wx = theta_w[c]; wy = theta_w[c + 1]; }
        else if (m == 1) { wx = g_w[c];     wy = g_w[c + 1];     }
        A[k].x = wx; A[k].y = wy;
    }

    const float* inb = in + (size_t)b * NC * NSP;
    float* gb = gbuf + (size_t)b * NSP;
    float t_part = 0.0f;

    for (int tile = waveId; tile < TILES_PER_BATCH; tile += K1_WAVES_PER_BATCH) {
        const int n = tile * 16 + m;
        const float* p = inb + (size_t)(2 * khalf) * NSP + n;
        v8f acc = {0.0f, 0.0f, 0.0f, 0.0f, 0.0f, 0.0f, 0.0f, 0.0f};
#pragma unroll
        for (int k = 0; k < 16; ++k) {
            v2f Bv;
            Bv.x = __builtin_nontemporal_load(p);        // channel 4k+2*khalf
            Bv.y = __builtin_nontemporal_load(p + NSP);  // channel 4k+2*khalf+1
            p += (size_t)4 * NSP;
            acc = __builtin_amdgcn_wmma_f32_16x16x4_f32(
                false, A[k], false, Bv, (short)0, acc, false, false);
        }
        t_part += acc[0];                 // row M=0 (lanes 16-31 carry M=8 == 0)
        if (lane < 16) gb[n] = acc[1];    // row M=1: g for 16 spatial positions
    }

    // Wave reduction of t, one atomic per wave per batch (8K atomics total).
#pragma unroll
    for (int off = 16; off > 0; off >>= 1)
        t_part += __shfl_xor(t_part, off, 32);
    if (lane == 0) atomicAdd(&tsum[b], t_part);
}

// ---------------------------------------------------------------------------
// Pass 2: out = in + g[b,n] * (tsum[b]/N) * h_w[c] + h_b[c], float4-vectorized
// streaming. in/out use non-temporal hints (read/written once); g stays
// L2-resident across its 64x reuse.
// ---------------------------------------------------------------------------
__global__ __launch_bounds__(256)
void k2_apply(const float* __restrict__ in,
              const float* __restrict__ gbuf,
              const float* __restrict__ tsum,
              const float* __restrict__ h_w,
              const float* __restrict__ h_b,
              float* __restrict__ out)
{
    const size_t f = (size_t)blockIdx.x * 256 + threadIdx.x;  // float4 index
    const int n4 = (int)(f & (NSP4 - 1));
    const int c  = (int)((f >> 15) & (NC - 1));
    const int b  = (int)(f >> 21);

    const v4f* in4 = (const v4f*)in;
    const v4f* g4  = (const v4f*)gbuf;

    v4f v = __builtin_nontemporal_load(&in4[f]);
    v4f g = g4[(size_t)b * NSP4 + n4];

    const float s  = tsum[b] * (1.0f / (float)NSP) * h_w[c];
    const float bb = h_b[c];

    v4f o;
    o.x = v.x + g.x * s + bb;
    o.y = v.y + g.y * s + bb;
    o.z = v.z + g.z * s + bb;
    o.w = v.w + g.w * s + bb;
    __builtin_nontemporal_store(o, &((v4f*)out)[f]);
}

extern "C" void kernel_launch(void* const* d_in, const int* in_sizes, int n_in,
                              void* d_out, int out_size, void* d_ws, size_t ws_size,
                              hipStream_t stream) {
    (void)in_sizes; (void)n_in; (void)out_size; (void)ws_size;

    const float* inpt    = (const float*)d_in[0];
    const float* theta_w = (const float*)d_in[1];
    const float* g_w     = (const float*)d_in[2];
    const float* h_w     = (const float*)d_in[3];
    const float* h_b     = (const float*)d_in[4];
    float* out = (float*)d_out;

    // Workspace layout: [0..7] t-sums per batch; g buffer at +256B (4 MB).
    float* tsum = (float*)d_ws;
    float* gbuf = (float*)((char*)d_ws + 256);

    hipMemsetAsync(tsum, 0, NB * sizeof(float), stream);  // capture-safe

    k1_contract_wmma<<<NB * K1_BLOCKS_PER_BATCH, 256, 0, stream>>>(
        inpt, theta_w, g_w, tsum, gbuf);

    const int total4 = NB * NC * NSP4;                    // 16,777,216
    k2_apply<<<total4 / 256, 256, 0, stream>>>(
        inpt, gbuf, tsum, h_w, h_b, out);
}